// CLShead_5712306504036
// MI455X (gfx1250) — compile-verified
//
#include <hip/hip_runtime.h>

typedef float v2f __attribute__((ext_vector_type(2)));
typedef float v4f __attribute__((ext_vector_type(4)));
typedef float v8f __attribute__((ext_vector_type(8)));

// ---- order-preserving float <-> u32 key (handles negatives, -0 < +0) ----
__device__ __forceinline__ unsigned flt_key(float f) {
    unsigned u = __float_as_uint(f);
    return (u & 0x80000000u) ? ~u : (u | 0x80000000u);
}
__device__ __forceinline__ float key_flt(unsigned u) {
    return (u & 0x80000000u) ? __uint_as_float(u & 0x7FFFFFFFu)
                             : __uint_as_float(~u);
}

// ---- zero the segment-max key buffer (lives in the M region of d_out) ----
__global__ void segmax_init_kernel(unsigned* __restrict__ keys, int Bn) {
    int i = blockIdx.x * blockDim.x + threadIdx.x;
    if (i < Bn) keys[i] = 0u;
}

// ---- main: WMMA f32 16x16x4 GEMV + fused segment-max ----
// One wave per 16-row tile. Tile staged to LDS coalesced, consumed in the
// documented WMMA A-layout (VGPR0: K0 | K2 across lane halves).
__global__ __launch_bounds__(128) void gemv_segmax_kernel(
    const float* __restrict__ z, const int* __restrict__ bag,
    const float* __restrict__ W, const float* __restrict__ pb,
    float* __restrict__ scores, unsigned* __restrict__ segkeys, int nTiles)
{
    __shared__ float smem[4 * 2048 + 128];   // 4 waves * 16x128 tile + W
    float* wlds = smem + 4 * 2048;

    const int tid  = threadIdx.x;
    const int lane = tid & 31;
    const int wave = tid >> 5;
    float* tileW = smem + wave * 2048;

    if (tid < 128) wlds[tid] = W[tid];       // stage W once per block
    __syncthreads();

    const float bias = pb[0];
    const int wavesTotal = (gridDim.x * blockDim.x) >> 5;
    const int gwave = (blockIdx.x * blockDim.x + tid) >> 5;

    const int m  = lane & 15;                // row within tile this lane feeds
    const int kk = (lane >> 4) << 1;         // lanes 0-15 -> K pair 0/1, 16-31 -> 2/3

    for (int tile = gwave; tile < nTiles; tile += wavesTotal) {
        const int row0 = tile << 4;

        // ---- stage 16x128 f32 tile: 512 contiguous bytes per b128 load ----
        const float* gsrc = z + (size_t)row0 * 128;
        #pragma unroll
        for (int it = 0; it < 16; ++it) {
            v4f v = *(const v4f*)(gsrc + it * 128 + lane * 4);
            *(v4f*)(tileW + it * 128 + lane * 4) = v;
        }
        // same-wave LDS ops are in-order; wait + fence for safety/ordering
        __asm__ volatile("s_wait_dscnt 0x0" ::: "memory");

        // ---- 32 x V_WMMA_F32_16X16X4_F32, K = 128 ----
        v8f acc = {0.f, 0.f, 0.f, 0.f, 0.f, 0.f, 0.f, 0.f};
        #pragma unroll
        for (int c = 0; c < 32; ++c) {
            const int k0 = c * 4;
            v2f av = *(const v2f*)(tileW + m * 128 + k0 + kk);  // A layout
            v2f bv = *(const v2f*)(wlds + k0 + kk);             // W broadcast to all N cols
            acc = __builtin_amdgcn_wmma_f32_16x16x4_f32(
                false, av, false, bv, (short)0, acc, false, false);
        }
        __asm__ volatile("" ::: "memory");

        // ---- epilogue: lane 0 holds rows 0-7 (col 0), lane 16 rows 8-15 ----
        if ((lane & 15) == 0) {
            const int rbase = row0 + ((lane >> 4) << 3);
            #pragma unroll
            for (int j = 0; j < 8; ++j) {
                float s = acc[j] + bias;
                scores[rbase + j] = s;
                int bg = bag[rbase + j];
                atomicMax(&segkeys[bg], flt_key(s));
            }
        }
    }
}

// ---- scalar tail for N % 16 rows (not hit for N = 500000) ----
__global__ void gemv_tail_kernel(
    const float* __restrict__ z, const int* __restrict__ bag,
    const float* __restrict__ W, const float* __restrict__ pb,
    float* __restrict__ scores, unsigned* __restrict__ segkeys,
    int startRow, int N)
{
    int r = startRow + blockIdx.x * blockDim.x + threadIdx.x;
    if (r >= N) return;
    float s = pb[0];
    for (int k = 0; k < 128; ++k) s += z[(size_t)r * 128 + k] * W[k];
    scores[r] = s;
    atomicMax(&segkeys[bag[r]], flt_key(s));
}

// ---- in-place: key buffer -> M floats; empty bag -> 0.0 ----
__global__ void segmax_final_kernel(unsigned* __restrict__ keys, int Bn) {
    int i = blockIdx.x * blockDim.x + threadIdx.x;
    if (i < Bn) {
        unsigned u = keys[i];
        ((float*)keys)[i] = (u == 0u) ? 0.0f : key_flt(u);
    }
}

extern "C" void kernel_launch(void* const* d_in, const int* in_sizes, int n_in,
                              void* d_out, int out_size, void* d_ws, size_t ws_size,
                              hipStream_t stream) {
    const float* z   = (const float*)d_in[0];   // (N,128) f32
    const int*   bag = (const int*)d_in[1];     // (N,)
    const float* W   = (const float*)d_in[2];   // (1,128) f32
    const float* pb  = (const float*)d_in[3];   // (1,) f32

    const int N  = in_sizes[1];
    const int Bn = out_size - N;                // 4096

    unsigned* segkeys = (unsigned*)d_out;       // M region doubles as key buffer
    float*    scores  = (float*)d_out + Bn;

    segmax_init_kernel<<<(Bn + 255) / 256, 256, 0, stream>>>(segkeys, Bn);

    const int nTiles = N >> 4;
    if (nTiles > 0) {
        gemv_segmax_kernel<<<2048, 128, 0, stream>>>(z, bag, W, pb,
                                                     scores, segkeys, nTiles);
    }
    const int rem = N & 15;
    if (rem > 0) {
        gemv_tail_kernel<<<(rem + 63) / 64, 64, 0, stream>>>(
            z, bag, W, pb, scores, segkeys, nTiles << 4, N);
    }
    segmax_final_kernel<<<(Bn + 255) / 256, 256, 0, stream>>>(segkeys, Bn);
}